// CombinedRotaryEmbedding_11381663334764
// MI455X (gfx1250) — compile-verified
//
#include <hip/hip_runtime.h>

typedef float v2f __attribute__((ext_vector_type(2)));
typedef float v8f __attribute__((ext_vector_type(8)));

#define H_DIM   64
#define NUM_ROT 32
#define S_LEN   8192   // sequence length from reference (B=4, S=8192)

// ---------------------------------------------------------------------------
// Prep: build the combined 64x64 matrix M' = (R0 R1 ... R31) * rotation_matrix
// with its COLUMNS permuted to [0,2,4,...,62,1,3,...,63] (folds the RoPE
// even/odd de-interleave into the GEMM), stored TRANSPOSED:
//   Mt[col][k]  (row-major 64x64)  so B-fragment loads are contiguous b64.
// One block of 64 threads; thread t owns permuted output column t.
// ---------------------------------------------------------------------------
__global__ void prep_combined_matrix(const float* __restrict__ thetas,
                                     const float* __restrict__ rot,
                                     float* __restrict__ Mt) {
  const int t = threadIdx.x;
  if (t >= H_DIM) return;
  const int outcol = (t < 32) ? (2 * t) : (2 * (t - 32) + 1);

  float v[H_DIM];
#pragma unroll
  for (int m = 0; m < H_DIM; ++m) v[m] = rot[m * H_DIM + outcol];

  // M[:,c] = R0 ( R1 ( ... ( R31 * rot[:,c] ) ) )  -> apply k = 31 .. 0
#pragma unroll
  for (int k = NUM_ROT - 1; k >= 0; --k) {
    const float c = cosf(thetas[k]);
    const float s = sinf(thetas[k]);
    const float vi = c * v[k] - s * v[k + 1];
    const float vj = s * v[k] + c * v[k + 1];
    v[k]     = vi;
    v[k + 1] = vj;
  }

#pragma unroll
  for (int m = 0; m < H_DIM; ++m) Mt[t * H_DIM + m] = v[m];
}

// ---------------------------------------------------------------------------
// Main: per wave, one (b,s) position = 16 head-rows x 64.
//   Y(16x64) = X(16x64) @ M'(64x64)   via 4 tiles x 16 WMMA_F32_16X16X4_F32
// B fragments (all of M') live in 128 VGPRs, loaded once per wave, reused
// across a grid-stride loop over positions. Epilogue = lane-local RoPE rotate
// between accumulator tiles {0,2} and {1,3} using 2 __sincosf per lane.
// ---------------------------------------------------------------------------
__global__ void __launch_bounds__(256)
rope_wmma_main(const float* __restrict__ x, const float* __restrict__ inv_freq,
               const float* __restrict__ Mt, float* __restrict__ out, int npos) {
  const int lane = threadIdx.x & 31;
  const int l16  = lane & 15;
  const int half = lane >> 4;                    // 0: K%4 in {0,1}, 1: {2,3}
  const int gwave  = (int)((blockIdx.x * blockDim.x + threadIdx.x) >> 5);
  const int nwaves = (int)((gridDim.x * blockDim.x) >> 5);

  // B fragments: bfrag[n][kk] holds M'[K][16n+l16] for K = 4kk+2*half, +1
  // Mt is stored transposed -> contiguous 8-byte load per fragment.
  v2f bfrag[4][16];
#pragma unroll
  for (int n = 0; n < 4; ++n)
#pragma unroll
    for (int kk = 0; kk < 16; ++kk)
      bfrag[n][kk] =
          *(const v2f*)(Mt + (16 * n + l16) * H_DIM + 4 * kk + 2 * half);

  const float f0 = inv_freq[l16];        // RoPE freq for cols  0..15 (tile 0/2)
  const float f1 = inv_freq[16 + l16];   // RoPE freq for cols 16..31 (tile 1/3)

  for (int pos = gwave; pos < npos; pos += nwaves) {
    const float* xt = x + (size_t)pos * 1024;    // 16 rows x 64 f32

    v8f acc0 = {}, acc1 = {}, acc2 = {}, acc3 = {};
#pragma unroll
    for (int kk = 0; kk < 16; ++kk) {
      // A fragment: row = l16, K = 4kk + 2*half + {0,1}  (contiguous b64)
      const v2f a = *(const v2f*)(xt + l16 * H_DIM + 4 * kk + 2 * half);
      acc0 = __builtin_amdgcn_wmma_f32_16x16x4_f32(false, a, false, bfrag[0][kk],
                                                   (short)0, acc0, false, false);
      acc1 = __builtin_amdgcn_wmma_f32_16x16x4_f32(false, a, false, bfrag[1][kk],
                                                   (short)0, acc1, false, false);
      acc2 = __builtin_amdgcn_wmma_f32_16x16x4_f32(false, a, false, bfrag[2][kk],
                                                   (short)0, acc2, false, false);
      acc3 = __builtin_amdgcn_wmma_f32_16x16x4_f32(false, a, false, bfrag[3][kk],
                                                   (short)0, acc3, false, false);
    }

    // RoPE epilogue: out[d] = y[d]*cos - y[d+32]*sin ; out[d+32] = y[d]*sin + y[d+32]*cos
    const int   s  = pos & (S_LEN - 1);
    const float fs = (float)s;
    float s0, c0, s1, c1;
    __sincosf(fs * f0, &s0, &c0);
    __sincosf(fs * f1, &s1, &c1);

    float* ot = out + (size_t)pos * 1024;
#pragma unroll
    for (int p = 0; p < 8; ++p) {
      const int row = p + 8 * half;              // C/D layout: VGPR p, lane half
      const float y0 = acc0[p], y1 = acc1[p], y2 = acc2[p], y3 = acc3[p];
      ot[row * H_DIM + 0  + l16] = y0 * c0 - y2 * s0;
      ot[row * H_DIM + 16 + l16] = y1 * c1 - y3 * s1;
      ot[row * H_DIM + 32 + l16] = y0 * s0 + y2 * c0;
      ot[row * H_DIM + 48 + l16] = y1 * s1 + y3 * c1;
    }
  }
}

extern "C" void kernel_launch(void* const* d_in, const int* in_sizes, int n_in,
                              void* d_out, int out_size, void* d_ws, size_t ws_size,
                              hipStream_t stream) {
  const float* x        = (const float*)d_in[0];
  const float* thetas   = (const float*)d_in[1];
  const float* rot      = (const float*)d_in[2];
  const float* inv_freq = (const float*)d_in[3];
  float*       out      = (float*)d_out;
  float*       Mt       = (float*)d_ws;          // 64*64 f32 = 16 KB scratch

  prep_combined_matrix<<<1, 64, 0, stream>>>(thetas, rot, Mt);

  const int npos = in_sizes[0] / 1024;           // (B*S) positions, 16 rows each
  int target_waves = (npos + 3) / 4;             // ~4 positions per wave
  int blocks = (target_waves + 7) / 8;           // 256 threads = 8 waves/block
  if (blocks < 1) blocks = 1;
  if (blocks > 4096) blocks = 4096;

  rope_wmma_main<<<blocks, 256, 0, stream>>>(x, inv_freq, Mt, out, npos);
}